// StochasticPool2d_5506148074043
// MI455X (gfx1250) — compile-verified
//
#include <hip/hip_runtime.h>

// Stochastic 2x2 pooling, B=32 C=64 H=256 W=256 -> out (B,C,128,128).
// Pure bandwidth-bound gather. Strategy: fully coalesced b128 streaming of both
// candidate x-rows + branchless register select; non-temporal hints everywhere
// (footprint ~0.8 GB >> 192 MB L2, zero reuse). HBM floor ~35 us @ 23.3 TB/s.

typedef float v4f __attribute__((ext_vector_type(4)));
typedef int   v4i __attribute__((ext_vector_type(4)));

#define W_FULL 256u   // x row length (floats)
#define W_OUT  128u   // out row length (floats)

__device__ __forceinline__ float sel4(float c0, float c1, float c2, float c3, int i) {
    // i in {0,1,2,3}: 0->c0 (r0,left) 1->c1 (r0,right) 2->c2 (r1,left) 3->c3 (r1,right)
    float lo = (i & 1) ? c1 : c0;
    float hi = (i & 1) ? c3 : c2;
    return (i & 2) ? hi : lo;
}

__global__ __launch_bounds__(256) void stochpool2x2_kernel(
    const float* __restrict__ x,
    const int*   __restrict__ idx,
    float*       __restrict__ out,
    unsigned n_out)              // total output elements
{
    const unsigned g = blockIdx.x * blockDim.x + threadIdx.x;
    if (g * 4u >= n_out) return;

    const unsigned lane = g & 31u;        // lane within wave32
    const unsigned row  = g >> 5;         // output row id: (b*C + c)*128 + ho
    const unsigned bc   = row >> 7;       // b*C + c
    const unsigned ho   = row & 127u;

    // x base: start of row pair (2*ho, 2*ho+1) in image bc, this lane's 8-col span
    const size_t xbase = ((size_t)bc * W_FULL + 2u * ho) * W_FULL + 8u * lane;
    const size_t obase = (size_t)row * W_OUT + 4u * lane;

    // Streaming, non-temporal b128 loads: idx (1), x top row (2), x bottom row (2)
    const v4i id  = __builtin_nontemporal_load((const v4i*)(idx + obase));
    const v4f r0a = __builtin_nontemporal_load((const v4f*)(x + xbase));
    const v4f r0b = __builtin_nontemporal_load((const v4f*)(x + xbase + 4u));
    const v4f r1a = __builtin_nontemporal_load((const v4f*)(x + xbase + W_FULL));
    const v4f r1b = __builtin_nontemporal_load((const v4f*)(x + xbase + W_FULL + 4u));

    // Branchless 1-of-4 select per output (3 v_cndmask each)
    v4f o;
    o.x = sel4(r0a.x, r0a.y, r1a.x, r1a.y, id.x);
    o.y = sel4(r0a.z, r0a.w, r1a.z, r1a.w, id.y);
    o.z = sel4(r0b.x, r0b.y, r1b.x, r1b.y, id.z);
    o.w = sel4(r0b.z, r0b.w, r1b.z, r1b.w, id.w);

    __builtin_nontemporal_store(o, (v4f*)(out + obase));
}

extern "C" void kernel_launch(void* const* d_in, const int* in_sizes, int n_in,
                              void* d_out, int out_size, void* d_ws, size_t ws_size,
                              hipStream_t stream) {
    const float* x   = (const float*)d_in[0];
    const int*   idx = (const int*)d_in[1];
    float*       out = (float*)d_out;

    const unsigned n_out    = (unsigned)out_size;          // 33,554,432
    const unsigned nthreads = (n_out + 3u) / 4u;           // 4 outputs per lane
    const unsigned block    = 256u;                        // 8 wave32s
    const unsigned grid     = (nthreads + block - 1u) / block;

    stochpool2x2_kernel<<<grid, block, 0, stream>>>(x, idx, out, n_out);
}